// KLDivergenceLoss_12463995093241
// MI455X (gfx1250) — compile-verified
//
#include <hip/hip_runtime.h>
#include <math.h>

// ---- problem constants (from reference setup_inputs) ----
#define B_        8
#define P_        200
#define H_        256
#define W_        256
#define C_        20
#define HW_       (H_ * W_)            // 65536
#define ROWS_     (B_ * P_)            // 1600 flat (b,p) planes
#define TILE_     16                   // hw pixels per tile
#define RS_       20                   // padded LDS row stride (floats): TDM pad 16+4
#define NT_       4                    // tiles per block (ping-pong pipeline)
#define THREADS_  128                  // 8 b  x 16 pix
#define NBLK_     (HW_ / (TILE_ * NT_))// 1024
#define IGNORE_L  250
#define EPS_      1e-8f

// LDS: 2 buffers [ROWS_][RS_] + per-buffer mmax/invZ [B_][TILE_]
#define BUF_FLOATS  (ROWS_ * RS_)                    // 32000
#define STAT_FLOATS (2 * B_ * TILE_)                 // mmax+invZ per buffer
#define LDS_FLOATS  (2 * BUF_FLOATS + 2 * STAT_FLOATS)

// ------------------------------------------------------------------
// Kernel 0: build per-class prototype lists from the identity matrix
// ------------------------------------------------------------------
__global__ void kl_setup_classes(const float* __restrict__ ident,
                                 int* __restrict__ kArr,
                                 int* __restrict__ plist) {
    int c = threadIdx.x;
    if (c < C_) {
        int k = 0;
        for (int p = 0; p < P_; ++p) {
            if (ident[p * C_ + c] == 1.0f) plist[c * P_ + (k++)] = p;
        }
        kArr[c] = k;
    }
}

// ------------------------------------------------------------------
// TDM descriptor issue: 2D tile (TILE_ floats x ROWS_ rows, row stride
// HW_ elements) DMA'd global -> LDS with pad-on-load (16 DW + 4 DW pad)
// ------------------------------------------------------------------
#if __has_builtin(__builtin_amdgcn_tensor_load_to_lds) && __has_builtin(__builtin_amdgcn_s_wait_tensorcnt)
#define HAVE_TDM 1
__device__ __forceinline__ void tdm_issue_tile(const float* gsrc, unsigned lds_off) {
    typedef unsigned int u32x4 __attribute__((ext_vector_type(4)));
    typedef int          i32x8 __attribute__((ext_vector_type(8)));
    typedef int          i32x4 __attribute__((ext_vector_type(4)));

    unsigned long long ga = (unsigned long long)(size_t)gsrc;

    u32x4 g0;
    g0.x = 1u;                                        // count=1
    g0.y = lds_off;                                   // LDS byte address
    g0.z = (unsigned)(ga & 0xFFFFFFFFull);            // global addr lo
    g0.w = (unsigned)((ga >> 32) & 0x01FFFFFFull) | 0x80000000u; // hi | type=2

    const unsigned long long td0 = (unsigned long long)HW_;   // tensor dim0
    const unsigned long long td1 = (unsigned long long)ROWS_; // tensor dim1
    const unsigned long long st0 = (unsigned long long)HW_;   // dim0 stride
    i32x8 g1;
    // data_size=4B (2<<16) | pad_enable (1<<20) |
    // pad_interval=16 DW (3<<22) | pad_amount=4 DW (3<<25)
    g1[0] = (int)((2u << 16) | (1u << 20) | (3u << 22) | (3u << 25));
    g1[1] = (int)((td0 & 0xFFFFull) << 16);                       // dim0[15:0]
    g1[2] = (int)(((td0 >> 16) & 0xFFFFull) |
                  ((td1 & 0xFFFFull) << 16));                     // dim0 hi | dim1 lo
    g1[3] = (int)(((td1 >> 16) & 0xFFFFull) |
                  ((unsigned long long)TILE_ << 16));             // dim1 hi | tile0
    g1[4] = (int)(ROWS_ & 0xFFFF);                                // tile1 | tile2=0
    g1[5] = (int)(st0 & 0xFFFFFFFFull);                           // stride0 lo
    g1[6] = (int)((st0 >> 32) & 0xFFFFull);                       // stride0 hi
    g1[7] = 0;

    i32x4 z4 = {0, 0, 0, 0};
    i32x8 z8 = {0, 0, 0, 0, 0, 0, 0, 0};
    __builtin_amdgcn_tensor_load_to_lds(g0, g1, z4, z4, z8, 0);
}
#else
#define HAVE_TDM 0
#endif

// ------------------------------------------------------------------
// Kernel 1: pipelined softmax + shuffled-KL.
// Each block processes NT_ tiles of TILE_ pixels; wave 0 keeps two TDM
// descriptors in flight (TENSORcnt in-order => s_wait_tensorcnt(1)
// guarantees the older DMA finished while the newer streams).
// ------------------------------------------------------------------
extern "C" __global__ void __launch_bounds__(THREADS_)
kl_main(const float* __restrict__ dist,
        const int*   __restrict__ labels,
        const int*   __restrict__ kArr,
        const int*   __restrict__ plist,
        float* __restrict__ blockTot,
        float* __restrict__ blockCnt) {
    extern __shared__ float smem[];
    float* bufs  = smem;                         // [2][ROWS_][RS_]
    float* stats = smem + 2 * BUF_FLOATS;        // [2][2][B_][TILE_]

    const int tid  = threadIdx.x;
    const int b    = tid >> 4;                   // 0..7
    const int pix  = tid & 15;                   // 0..15
    const int hwB  = blockIdx.x * (TILE_ * NT_);

#if HAVE_TDM
    typedef __attribute__((address_space(3))) float lds_f;
    unsigned lds_base = (unsigned)(size_t)(lds_f*)bufs;
    if (tid < 32) {                              // wave 0 drives the DMA pipeline
        tdm_issue_tile(dist + hwB + 0 * TILE_, lds_base + 0 * BUF_FLOATS * 4);
        tdm_issue_tile(dist + hwB + 1 * TILE_, lds_base + 1 * BUF_FLOATS * 4);
    }
#endif

    float accP = 0.f, accC = 0.f;                // running loss / count

    for (int t = 0; t < NT_; ++t) {
        float* raw  = bufs + (t & 1) * BUF_FLOATS;
        float* mmax = stats + (t & 1) * STAT_FLOATS;
        float* invZ = mmax + B_ * TILE_;
        const int hw0 = hwB + t * TILE_;

#if HAVE_TDM
        if (tid < 32) {
            if (t < NT_ - 1) __builtin_amdgcn_s_wait_tensorcnt(1); // tile t landed
            else             __builtin_amdgcn_s_wait_tensorcnt(0);
        }
#else
        // Fallback: coalesced float4 copy into padded rows
        for (int i = 0; i < (ROWS_ * TILE_) / (4 * THREADS_); ++i) {
            int j   = i * THREADS_ + tid;
            int row = j >> 2;                    // 4 x 16B chunks per 64B row
            int sub = (j & 3) * 4;
            const float4 v = *reinterpret_cast<const float4*>(
                dist + (size_t)row * HW_ + hw0 + sub);
            *reinterpret_cast<float4*>(raw + row * RS_ + sub) = v;
        }
#endif
        __syncthreads();

        // ---- Phase B: softmax stats over P for this thread's (b,pix) ----
        {
            const float* rb = raw + (size_t)(b * P_) * RS_ + pix;
            float mx = -3.402823466e38f;
#pragma unroll 4
            for (int p = 0; p < P_; ++p) mx = fmaxf(mx, -rb[p * RS_]);
            float s = 0.f;
#pragma unroll 4
            for (int p = 0; p < P_; ++p) s += __expf(-rb[p * RS_] - mx);
            mmax[b * TILE_ + pix] = mx;
            invZ[b * TILE_ + pix] = 1.0f / s;
        }
        __syncthreads();

        // ---- Phase C: shuffled-KL term for loss pixel (b, hw0+pix) ----
        {
            int lab = labels[(size_t)b * HW_ + hw0 + pix];
            bool valid = (lab != IGNORE_L);
            int c = min(max(lab, 0), C_ - 1);
            int k = kArr[c];
            if (valid && k > 0) {
                float a = 0.f;
                for (int i = 0; i < k; ++i) {
                    int p = plist[c * P_ + i];
                    int m = p * B_ + b;          // flat plane index == b'*P + p'
                    int b2 = m / P_;
                    float x   = -raw[m * RS_ + pix];
                    float act = __expf(x - mmax[b2 * TILE_ + pix]) *
                                invZ[b2 * TILE_ + pix];
                    a += __logf(act + EPS_);
                }
                float kk = (float)k;
                accP += -__logf(kk) - a / kk;
                accC += 1.f;
            }
        }
        __syncthreads();                         // buffer free before re-issue

#if HAVE_TDM
        if (tid < 32 && (t + 2) < NT_) {
            tdm_issue_tile(dist + hwB + (t + 2) * TILE_,
                           lds_base + (t & 1) * BUF_FLOATS * 4);
        }
#endif
    }

    // deterministic block reduction: shuffle within wave32, LDS across waves
    for (int o = 16; o > 0; o >>= 1) {
        accP += __shfl_down(accP, o, 32);
        accC += __shfl_down(accC, o, 32);
    }
    __syncthreads();                             // buffers done: reuse as scratch
    if ((tid & 31) == 0) {
        bufs[tid >> 5]              = accP;
        bufs[8 + (tid >> 5)]        = accC;
    }
    __syncthreads();
    if (tid == 0) {
        float tp = 0.f, tc = 0.f;
        for (int w = 0; w < THREADS_ / 32; ++w) { tp += bufs[w]; tc += bufs[8 + w]; }
        blockTot[blockIdx.x] = tp;
        blockCnt[blockIdx.x] = tc;
    }
}

// ------------------------------------------------------------------
// Kernel 2: fixed-order final reduction -> scalar mean
// ------------------------------------------------------------------
__global__ void kl_finalize(const float* __restrict__ bt,
                            const float* __restrict__ bc,
                            float* __restrict__ out) {
    __shared__ float s1[256];
    __shared__ float s2[256];
    int t = threadIdx.x;
    float a = 0.f, b = 0.f;
    for (int i = t; i < NBLK_; i += 256) { a += bt[i]; b += bc[i]; }
    s1[t] = a; s2[t] = b;
    __syncthreads();
    for (int o = 128; o > 0; o >>= 1) {
        if (t < o) { s1[t] += s1[t + o]; s2[t] += s2[t + o]; }
        __syncthreads();
    }
    if (t == 0) out[0] = (s2[0] > 0.f) ? s1[0] / fmaxf(s2[0], 1.f) : 0.f;
}

// ------------------------------------------------------------------
extern "C" void kernel_launch(void* const* d_in, const int* in_sizes, int n_in,
                              void* d_out, int out_size, void* d_ws, size_t ws_size,
                              hipStream_t stream) {
    const float* dist   = (const float*)d_in[0];   // [B,P,H,W] f32
    const float* ident  = (const float*)d_in[1];   // [P,C] f32
    const int*   labels = (const int*)d_in[2];     // [B,H,W] i32
    float* out = (float*)d_out;

    // workspace layout
    int*   kArr     = (int*)d_ws;
    int*   plist    = kArr + C_;
    float* blockTot = (float*)(plist + C_ * P_);
    float* blockCnt = blockTot + NBLK_;

    kl_setup_classes<<<1, 32, 0, stream>>>(ident, kArr, plist);

    size_t shmem = (size_t)LDS_FLOATS * sizeof(float);   // ~258 KB < 320 KB/WGP
    kl_main<<<NBLK_, THREADS_, shmem, stream>>>(dist, labels, kArr, plist,
                                                blockTot, blockCnt);

    kl_finalize<<<1, 256, 0, stream>>>(blockTot, blockCnt, out);
}